// VQRNonLinearModel_18605798326519
// MI455X (gfx1250) — compile-verified
//
#include <hip/hip_runtime.h>

// ---------------------------------------------------------------------------
// 16-qubit VQC simulator for MI455X (gfx1250, wave32).
//   - 4 single-qubit gates per nibble fused into 16x16 complex unitaries
//   - applied as complex GEMMs with V_WMMA_F32_16X16X4_F32 (fp32 precision)
//   - CNOT ring = GF(2)-linear bit permutation, fused into store addressing
// Workspace layout (requires ~528 MB):
//   [0, 256MB)      state buffer A (512 x 65536 float2)
//   [256MB, 512MB)  state buffer B
//   [512MB, 528MB)  fused gate matrices: [layer][nibble][batch][16][16] float2
// ---------------------------------------------------------------------------

typedef float v2f __attribute__((ext_vector_type(2)));
typedef float v8f __attribute__((ext_vector_type(8)));

__device__ __forceinline__ v8f wmma_f32(v2f a, v2f b, v8f c) {
  // D = A(16x4) * B(4x16) + C(16x16), fp32, wave32
  return __builtin_amdgcn_wmma_f32_16x16x4_f32(false, a, false, b, (short)0, c,
                                               false, false);
}

struct CFrag { v8f re; v8f im; };

// 16x16x16 complex GEMM: C = A * B.
// A: fused gate matrix, row-major float2[16][16] in global (L2-resident).
// B: state tile in LDS planes (re/im), element [k][n] at k*17+n (pad 17).
// Result left in WMMA C layout (VGPR v, lane<16 -> M=v, lane>=16 -> M=v+8).
__device__ __forceinline__ CFrag cgemm16(const float2* __restrict__ A,
                                         const float* sRe, const float* sIm,
                                         int lane) {
  v8f cre = {0.f, 0.f, 0.f, 0.f, 0.f, 0.f, 0.f, 0.f};
  v8f cim = cre;
  const int row = lane & 15;  // A-row (M); also B-col (N) for this lane
  const int hf = lane >> 4;
#pragma unroll
  for (int kb = 0; kb < 4; ++kb) {
    const int k0 = kb * 4 + hf * 2;
    float2 a0 = A[row * 16 + k0];
    float2 a1 = A[row * 16 + k0 + 1];
    v2f ar = {a0.x, a1.x};
    v2f ai = {a0.y, a1.y};
    v2f an = {-a0.y, -a1.y};  // -Im(A) for the real-part accumulation
    v2f br = {sRe[k0 * 17 + row], sRe[(k0 + 1) * 17 + row]};
    v2f bi = {sIm[k0 * 17 + row], sIm[(k0 + 1) * 17 + row]};
    cre = wmma_f32(ar, br, cre);
    cre = wmma_f32(an, bi, cre);
    cim = wmma_f32(ar, bi, cim);
    cim = wmma_f32(ai, br, cim);
  }
  CFrag r; r.re = cre; r.im = cim; return r;
}

__device__ __forceinline__ float2 cmul(float2 a, float2 b) {
  return make_float2(a.x * b.x - a.y * b.y, a.x * b.y + a.y * b.x);
}

// CNOT-ring permutation of a 16-bit basis index (suffix-XOR scan + wrap CNOT).
__device__ __forceinline__ unsigned cnot_ring_perm(unsigned m) {
  unsigned y = m;
  y ^= y >> 1; y ^= y >> 2; y ^= y >> 4; y ^= y >> 8;  // y_i = XOR of bits i..15
  unsigned b15 = ((m >> 15) ^ (y & 1u)) & 1u;          // final CNOT: b15 ^= b0'
  return (y & 0x7FFFu) | (b15 << 15);
}

// Build fused 16x16 complex gate matrix per (layer, nibble, batch).
// Nibble g covers index bits [4g+3 : 4g]; nibble-bit k <-> wire (15-4g-k).
__global__ __launch_bounds__(256) void k_mats(const float* __restrict__ X,
                                              const float* __restrict__ W,
                                              float2* __restrict__ mats) {
  const int lg = blockIdx.x >> 9;  // l*4 + g
  const int b = blockIdx.x & 511;
  const int l = lg >> 2, g = lg & 3;
  const int tid = threadIdx.x;
  const int i = tid >> 4, j = tid & 15;
  float2 prod = make_float2(1.f, 0.f);
#pragma unroll
  for (int k = 0; k < 4; ++k) {
    const int w = 15 - 4 * g - k;
    // RX(2*atan(x)): cos(phi/2)=1/sqrt(1+x^2), sin(phi/2)=x/sqrt(1+x^2)
    float x = X[b * 16 + w];
    float cr = rsqrtf(1.f + x * x);
    float sr = x * cr;
    float2 u00 = make_float2(cr, 0.f),  u01 = make_float2(0.f, -sr);
    float2 u10 = make_float2(0.f, -sr), u11 = make_float2(cr, 0.f);
    const float* th = W + (l * 16 + w) * 3;
    {  // RZ(t0) left-mul: rows scaled by e^{-+i t/2}
      float h = 0.5f * th[0], c = __cosf(h), s = __sinf(h);
      float2 e0 = make_float2(c, -s), e1 = make_float2(c, s);
      u00 = cmul(e0, u00); u01 = cmul(e0, u01);
      u10 = cmul(e1, u10); u11 = cmul(e1, u11);
    }
    {  // RY(t1) left-mul: [[c,-s],[s,c]] (real)
      float h = 0.5f * th[1], c = __cosf(h), s = __sinf(h);
      float2 r00 = u00, r01 = u01;
      u00 = make_float2(c * r00.x - s * u10.x, c * r00.y - s * u10.y);
      u01 = make_float2(c * r01.x - s * u11.x, c * r01.y - s * u11.y);
      u10 = make_float2(s * r00.x + c * u10.x, s * r00.y + c * u10.y);
      u11 = make_float2(s * r01.x + c * u11.x, s * r01.y + c * u11.y);
    }
    {  // RZ(t2) left-mul
      float h = 0.5f * th[2], c = __cosf(h), s = __sinf(h);
      float2 e0 = make_float2(c, -s), e1 = make_float2(c, s);
      u00 = cmul(e0, u00); u01 = cmul(e0, u01);
      u10 = cmul(e1, u10); u11 = cmul(e1, u11);
    }
    const int ik = (i >> k) & 1, jk = (j >> k) & 1;
    float2 sel = ik ? (jk ? u11 : u10) : (jk ? u01 : u00);
    prod = cmul(prod, sel);
  }
  mats[(size_t)blockIdx.x * 256 + tid] = prod;
}

__global__ __launch_bounds__(256) void k_init(float2* __restrict__ buf) {
  long long g = (long long)blockIdx.x * 256 + threadIdx.x;
  buf[g] = make_float2((g & 65535) == 0 ? 1.f : 0.f, 0.f);
}

// Pass over bits 7..0: chunk = 256 contiguous amplitudes (fixed b, bits15-8).
// New = G1 * S * G0^T via two WMMA complex GEMMs with an LDS transpose. In-place.
__global__ __launch_bounds__(256) void k_gates_lo(float2* __restrict__ buf,
                                                  const float2* __restrict__ mG1,
                                                  const float2* __restrict__ mG0) {
  __shared__ float lds[8 * 544];  // per-wave slab: re[16][17] + im[16][17]
  const int tid = threadIdx.x, lane = tid & 31, wv = tid >> 5;
  float* sRe = lds + wv * 544;
  float* sIm = sRe + 272;
  const long long chunk = (long long)blockIdx.x * 8 + wv;
  const int b = (int)(chunk >> 8);
  const int hi8 = (int)(chunk & 255);
  float2* base = buf + ((long long)b << 16) + hi8 * 256;
  const float2* G1 = mG1 + (size_t)b * 256;
  const float2* G0 = mG0 + (size_t)b * 256;
#pragma unroll
  for (int t = 0; t < 8; ++t) {  // coalesced 256-elem tile load -> LDS planes
    int e = lane + t * 32;
    float2 v = base[e];
    sRe[(e >> 4) * 17 + (e & 15)] = v.x;
    sIm[(e >> 4) * 17 + (e & 15)] = v.y;
  }
  __syncthreads();
  CFrag T = cgemm16(G1, sRe, sIm, lane);  // T = G1 * S
  __syncthreads();
  const int n = lane & 15, hf = lane >> 4;
#pragma unroll
  for (int v = 0; v < 8; ++v) {  // store T^T back to LDS
    int m = v + 8 * hf;
    sRe[n * 17 + m] = T.re[v];
    sIm[n * 17 + m] = T.im[v];
  }
  __syncthreads();
  CFrag O = cgemm16(G0, sRe, sIm, lane);  // O = G0 * T^T  (= New^T)
#pragma unroll
  for (int v = 0; v < 8; ++v) {  // O[i][r] = New[r][i]; elem = r*16 + i
    int i = v + 8 * hf;
    base[n * 16 + i] = make_float2(O.re[v], O.im[v]);
  }
}

// Pass over bits 15..8 (stride-256 tiles) + CNOT-ring permutation fused into
// the scatter store. Reads src, writes dst (ping-pong).
__global__ __launch_bounds__(256) void k_gates_hi(const float2* __restrict__ src,
                                                  float2* __restrict__ dst,
                                                  const float2* __restrict__ mG3,
                                                  const float2* __restrict__ mG2) {
  __shared__ float lds[8 * 544];
  const int tid = threadIdx.x, lane = tid & 31, wv = tid >> 5;
  const int blk = blockIdx.x;
  const int b = blk >> 5;
  const int L = (blk & 31) * 8;  // lo8 base for this block's 8 chunks
  const float2* sbase = src + ((long long)b << 16) + L;
  for (int t = tid; t < 2048; t += 256) {  // cooperative strided-tile load
    int h = t >> 3, s = t & 7;
    float2 v = sbase[(long long)h * 256 + s];
    float* r = lds + s * 544;
    r[(h >> 4) * 17 + (h & 15)] = v.x;
    (r + 272)[(h >> 4) * 17 + (h & 15)] = v.y;
  }
  __syncthreads();
  float* sRe = lds + wv * 544;
  float* sIm = sRe + 272;
  const float2* G3 = mG3 + (size_t)b * 256;
  const float2* G2 = mG2 + (size_t)b * 256;
  CFrag T = cgemm16(G3, sRe, sIm, lane);  // gate on bits 15-12
  __syncthreads();
  const int n = lane & 15, hf = lane >> 4;
#pragma unroll
  for (int v = 0; v < 8; ++v) {
    int m = v + 8 * hf;
    sRe[n * 17 + m] = T.re[v];
    sIm[n * 17 + m] = T.im[v];
  }
  __syncthreads();
  CFrag O = cgemm16(G2, sRe, sIm, lane);  // gate on bits 11-8
  const int lo8 = L + wv;
  float2* dbase = dst + ((long long)b << 16);
#pragma unroll
  for (int v = 0; v < 8; ++v) {
    int i2 = v + 8 * hf;                               // bits 11-8
    unsigned m = ((unsigned)(n * 16 + i2) << 8) | lo8; // n = bits 15-12
    dbase[cnot_ring_perm(m)] = make_float2(O.re[v], O.im[v]);
  }
}

// z[b] = sum_{bit15=0}|amp|^2 - sum_{bit15=1}|amp|^2 ; out = z + bias
__global__ __launch_bounds__(256) void k_out(const float2* __restrict__ st,
                                             const float* __restrict__ bias,
                                             float* __restrict__ out) {
  __shared__ float red[256];
  const int b = blockIdx.x, tid = threadIdx.x;
  const float2* p = st + ((long long)b << 16);
  float acc = 0.f;
  for (int k = 0; k < 256; ++k) {
    float2 v = p[(long long)k * 256 + tid];
    float pr = v.x * v.x + v.y * v.y;
    acc += (k < 128) ? pr : -pr;
  }
  red[tid] = acc;
  __syncthreads();
  for (int s = 128; s > 0; s >>= 1) {
    if (tid < s) red[tid] += red[tid + s];
    __syncthreads();
  }
  if (tid == 0) out[b] = red[0] + bias[0];
}

extern "C" void kernel_launch(void* const* d_in, const int* in_sizes, int n_in,
                              void* d_out, int out_size, void* d_ws,
                              size_t ws_size, hipStream_t stream) {
  const float* X = (const float*)d_in[0];     // 512 x 16
  const float* W = (const float*)d_in[1];     // 4 x 16 x 3
  const float* bias = (const float*)d_in[2];  // 1
  float* out = (float*)d_out;                 // 512

  char* ws = (char*)d_ws;
  const size_t STATE_BYTES = (size_t)512 * 65536 * sizeof(float2);  // 256 MB
  float2* buf0 = (float2*)ws;
  float2* buf1 = (float2*)(ws + STATE_BYTES);
  float2* mats = (float2*)(ws + 2 * STATE_BYTES);  // 16 MB

  k_mats<<<8192, 256, 0, stream>>>(X, W, mats);
  k_init<<<131072, 256, 0, stream>>>(buf0);

  float2* cur = buf0;
  float2* nxt = buf1;
  for (int l = 0; l < 4; ++l) {
    const float2* mg = mats + (size_t)(l * 4) * 512 * 256;
    k_gates_lo<<<16384, 256, 0, stream>>>(cur, mg + (size_t)1 * 512 * 256,
                                          mg + (size_t)0 * 512 * 256);
    k_gates_hi<<<16384, 256, 0, stream>>>(cur, nxt, mg + (size_t)3 * 512 * 256,
                                          mg + (size_t)2 * 512 * 256);
    float2* t = cur; cur = nxt; nxt = t;
  }
  k_out<<<512, 256, 0, stream>>>(cur, bias, out);
}